// DGLLayer_18502719111841
// MI455X (gfx1250) — compile-verified
//
#include <hip/hip_runtime.h>

typedef __attribute__((ext_vector_type(2))) float v2f;
typedef __attribute__((ext_vector_type(8))) float v8f;

#define N_DRUG 25000
#define N_DIS  25000
#define N_NODE 50000
#define DFEAT  128
#define E_EDGE 800000
#define ROWBLKS ((N_DRUG + 15) / 16)   // 1563 (same for disease)

// ---------------------------------------------------------------- utilities
__global__ void zero_f32_kernel(float* __restrict__ p, int n) {
    int i = blockIdx.x * blockDim.x + threadIdx.x;
    int stride = gridDim.x * blockDim.x;
    for (; i < n; i += stride) p[i] = 0.0f;
}

__global__ void degree_kernel(const int* __restrict__ rows, float* __restrict__ deg) {
    int e = blockIdx.x * blockDim.x + threadIdx.x;
    if (e < E_EDGE) atomicAdd(&deg[rows[e]], 1.0f);
}

// ------------------------------------------------- fp32 WMMA projection
// One block = one 16-row tile of one node type; 8 waves cover the 8 column
// tiles. A-tile (16x128 f32, 8KB) is staged once in LDS (coalesced, with
// zero-fill for the ragged last block) so the K-loop is branchless:
// 32 chained V_WMMA_F32_16X16X4_F32 per wave, A from ds_load, B from L2.
// Epilogue fuses the cols-side symmetric norm: h[i] *= rsqrt(max(deg[i],1)).
__global__ __launch_bounds__(256)
void project_wmma_kernel(const float* __restrict__ drugF,
                         const float* __restrict__ disF,
                         const float* __restrict__ drugW,
                         const float* __restrict__ disW,
                         const float* __restrict__ deg,
                         float* __restrict__ nodeH) {
    __shared__ float As[16 * DFEAT];         // 8 KB of the 320 KB/WGP pool

    const int blk     = blockIdx.x;
    const bool isDrug = (blk < ROWBLKS);
    const int rowBlk  = isDrug ? blk : blk - ROWBLKS;
    const float* __restrict__ A = isDrug ? drugF : disF;
    const float* __restrict__ W = isDrug ? drugW : disW;
    const int nRows   = isDrug ? N_DRUG : N_DIS;
    const int gBase   = isDrug ? 0 : N_DRUG;
    const int rowBase = rowBlk * 16;

    // ---- stage A tile: 512 float4s, 2 per thread, bounds handled here only
    #pragma unroll
    for (int i = 0; i < 2; ++i) {
        const int idx      = threadIdx.x + i * 256;   // float4 index [0,512)
        const int rowLocal = idx >> 5;                // 32 float4 per row
        const int colQ     = idx & 31;
        const int grow     = rowBase + rowLocal;
        float4 v = make_float4(0.f, 0.f, 0.f, 0.f);
        if (grow < nRows)
            v = ((const float4*)(A + (long long)grow * DFEAT))[colQ];
        ((float4*)As)[idx] = v;
    }
    __syncthreads();

    const int wave    = threadIdx.x >> 5;    // 8 waves -> 8 column tiles
    const int lane    = threadIdx.x & 31;
    const int colBase = wave * 16;
    const int m       = lane & 15;           // row (A) / col (B,C) within tile
    const int h       = lane >> 4;           // lane half selects K pair
    const float* __restrict__ aLds = As + m * DFEAT;
    const float* __restrict__ wCol = W + colBase + m;

    v8f c = {};
    #pragma unroll 8
    for (int k4 = 0; k4 < DFEAT; k4 += 4) {
        const int kb = k4 + 2 * h;           // lane half h holds K=kb, kb+1
        v2f a, b;
        const float2 av = *(const float2*)(aLds + kb);        // ds_load_b64
        a.x = av.x; a.y = av.y;
        b.x = wCol[(long long)kb * DFEAT];                    // W[kb][col]
        b.y = wCol[(long long)(kb + 1) * DFEAT];              // W[kb+1][col]
        c = __builtin_amdgcn_wmma_f32_16x16x4_f32(
                /*neg_a=*/false, a, /*neg_b=*/false, b,
                /*c_mod=*/(short)0, c, /*reuse_a=*/false, /*reuse_b=*/false);
    }

    // Epilogue: C VGPR v -> row (v + 8h), col m; scale by rsqrt(max(deg,1)).
    #pragma unroll
    for (int v = 0; v < 8; ++v) {
        const int r = rowBase + v + 8 * h;
        if (r < nRows) {
            const int g = gBase + r;
            const float d = deg[g];
            const float nrm = __frsqrt_rn(d < 1.0f ? 1.0f : d);
            nodeH[(long long)g * DFEAT + colBase + m] = c[v] * nrm;
        }
    }
}

// ------------------------------------------------- SpMM scatter (wave/edge)
// One wave per edge: lanes cooperatively gather the full 512B source row
// (float4 each, L2-resident) and scatter-add the destination row with
// global_atomic_add_f32.
__global__ __launch_bounds__(256)
void spmm_scatter_kernel(const int* __restrict__ rows,
                         const int* __restrict__ cols,
                         const float* __restrict__ nodeH,
                         float* __restrict__ out) {
    const long long t = (long long)blockIdx.x * blockDim.x + threadIdx.x;
    const int e = (int)(t >> 5);             // 32 lanes per edge
    if (e >= E_EDGE) return;
    const int l = (int)(t & 31);
    const int r = rows[e];
    const int c = cols[e];
    const float4 src = ((const float4*)(nodeH + (long long)c * DFEAT))[l];
    float* dst = out + (long long)r * DFEAT + l * 4;
    atomicAdd(dst + 0, src.x);
    atomicAdd(dst + 1, src.y);
    atomicAdd(dst + 2, src.z);
    atomicAdd(dst + 3, src.w);
}

// ------------------------------------------------- rows-side norm epilogue
__global__ void scale_out_kernel(float* __restrict__ out,
                                 const float* __restrict__ deg) {
    const int i = blockIdx.x * blockDim.x + threadIdx.x;
    if (i < N_NODE * DFEAT) {
        const float d = deg[i >> 7];
        out[i] *= __frsqrt_rn(d < 1.0f ? 1.0f : d);
    }
}

// ---------------------------------------------------------------- launcher
extern "C" void kernel_launch(void* const* d_in, const int* in_sizes, int n_in,
                              void* d_out, int out_size, void* d_ws, size_t ws_size,
                              hipStream_t stream) {
    const float* drug_f = (const float*)d_in[0];
    const float* dis_f  = (const float*)d_in[1];
    const float* drug_w = (const float*)d_in[2];
    const float* dis_w  = (const float*)d_in[3];
    const int*   rows   = (const int*)d_in[4];
    const int*   cols   = (const int*)d_in[5];
    float* out = (float*)d_out;

    float* deg   = (float*)d_ws;             // 50000 floats
    float* nodeH = deg + 51200;              // 256B-aligned, 50000*128 floats

    // 1) zero degree histogram and output accumulator
    zero_f32_kernel<<<256, 256, 0, stream>>>(deg, N_NODE);
    zero_f32_kernel<<<4096, 256, 0, stream>>>(out, N_NODE * DFEAT);
    // 2) degrees
    degree_kernel<<<(E_EDGE + 255) / 256, 256, 0, stream>>>(rows, deg);
    // 3) WMMA projection + cols-side norm  (2 types x 1563 row blocks)
    project_wmma_kernel<<<2 * ROWBLKS, 256, 0, stream>>>(
        drug_f, dis_f, drug_w, dis_w, deg, nodeH);
    // 4) edge scatter: E*32 threads = 25.6M -> 100000 blocks exactly
    spmm_scatter_kernel<<<(int)(((long long)E_EDGE * 32) / 256), 256, 0, stream>>>(
        rows, cols, nodeH, out);
    // 5) rows-side norm
    scale_out_kernel<<<(N_NODE * DFEAT + 255) / 256, 256, 0, stream>>>(out, deg);
}